// PureTriadicBrain_73753178407515
// MI455X (gfx1250) — compile-verified
//
#include <hip/hip_runtime.h>
#include <math.h>

// MI455X / gfx1250, wave32. WMMA F32 16x16x4 matches HD=4 exactly.

typedef __attribute__((ext_vector_type(2))) float v2f;
typedef __attribute__((ext_vector_type(8))) float v8f;

#define B_ 4
#define N_ 512
#define D_ 64
#define H_ 16
#define HD_ 4
#define EPS_ 1e-8f
#define GAIN_ 1.8477590650225735f   // sqrt(2 + sqrt(2))
#define SINK_ITERS 5

__device__ __forceinline__ float wave_reduce_sum(float v) {
  #pragma unroll
  for (int off = 16; off > 0; off >>= 1)
    v += __shfl_xor(v, off, 32);
  return v;
}

// ---- zero the per-(b,h) sum-of-squares accumulators ----
__global__ void k_init(float* ss) {
  if (threadIdx.x < B_ * H_) ss[threadIdx.x] = 0.0f;
}

// ---- queries/keys projection: per-node 64x64 matvec + buntanh ----
// (the `prediction` / w1 branch of the reference is dead code -> skipped)
__global__ void k_proj(const float* __restrict__ state, const float* __restrict__ w2,
                       const float* __restrict__ w3, const float* __restrict__ hp_u,
                       float* __restrict__ Q, float* __restrict__ K) {
  const int bn = blockIdx.x;               // 0..B*N-1
  const int b = bn / N_, n = bn % N_;
  const int e = threadIdx.x;               // 0..63 (output column)
  __shared__ float s[D_];
  s[e] = state[(size_t)bn * D_ + e];
  __syncthreads();
  const float* w2p = w2 + (size_t)bn * D_ * D_;
  const float* w3p = w3 + (size_t)bn * D_ * D_;
  float aq = 0.f, ak = 0.f;
  #pragma unroll 8
  for (int d = 0; d < D_; ++d) {           // coalesced: 64 lanes read row d
    const float sv = s[d];
    aq += sv * w3p[d * D_ + e];
    ak += sv * w2p[d * D_ + e];
  }
  const float u  = hp_u[b];
  const float qv = GAIN_ * tanhf(aq * u) / u;
  const float kv = GAIN_ * tanhf(ak * u) / u;
  // store as (B,H,N,HD) so attention kernels read contiguous float2/float4
  const int h = e >> 2, hd = e & 3;
  const size_t qi = (((size_t)(b * H_ + h) * N_) + n) * HD_ + hd;
  Q[qi] = qv;
  K[qi] = kv;
}

// ---- raw_A = Q K^T / sqrt(HD) via V_WMMA_F32_16X16X4_F32; also sum(x^2) per slice ----
// one wave per (bh, 16-row strip); loops 32 column tiles; A-operand (Q tile) reused.
__global__ void k_rawA(const float* __restrict__ Q, const float* __restrict__ K,
                       float* __restrict__ rawA, float* __restrict__ ss) {
  const int lane = threadIdx.x & 31;
  const int wid  = blockIdx.x * (blockDim.x >> 5) + (threadIdx.x >> 5);
  const int bh = wid >> 5;                 // 0..63
  const int nt = wid & 31;                 // n-tile (16 rows)
  const int r16 = lane & 15;
  const int k0  = (lane >> 4) << 1;        // ISA 16x4 f32 A layout: K = vgpr + 2*(lane/16)
  const v2f a = *(const v2f*)(Q + ((size_t)bh * N_ + nt * 16 + r16) * HD_ + k0);
  float ssa = 0.f;
  for (int mt = 0; mt < 32; ++mt) {
    const v2f bm = *(const v2f*)(K + ((size_t)bh * N_ + mt * 16 + r16) * HD_ + k0);
    v8f c = {};
    c = __builtin_amdgcn_wmma_f32_16x16x4_f32(false, a, false, bm, (short)0, c, false, false);
    const int rowbase = nt * 16 + ((lane >> 4) << 3);   // D layout: M = vgpr + 8*(lane/16)
    float* dst = rawA + (size_t)bh * N_ * N_ + mt * 16 + r16;
    #pragma unroll
    for (int r = 0; r < 8; ++r) {
      const float v = c[r] * 0.5f;                      // 1/sqrt(HD)
      dst[(size_t)(rowbase + r) * N_] = v;
      ssa += v * v;
    }
  }
  ssa = wave_reduce_sum(ssa);
  if (lane == 0) atomicAdd(&ss[bh], ssa);
}

// ---- rms_norm (per-slice) + EMA + input-node row mask, elementwise in place ----
__global__ void k_ema(float* __restrict__ A, const float* __restrict__ ema,
                      const float* __restrict__ ss, const float* __restrict__ hp_cm,
                      const float* __restrict__ stepc, int nin) {
  const size_t total = (size_t)B_ * H_ * N_ * N_;
  const float isf = (stepc[0] == 1.0f) ? 1.f : 0.f;
  for (size_t i = (size_t)blockIdx.x * blockDim.x + threadIdx.x; i < total;
       i += (size_t)gridDim.x * blockDim.x) {
    const int bh = (int)(i >> 18);                 // N*N == 2^18
    const int b  = bh >> 4;
    const int n  = (int)((i >> 9) & (N_ - 1));
    const float scale = rsqrtf(ss[bh] * (1.f / ((float)N_ * N_)) + EPS_);
    const float raw = A[i] * scale;
    const float cm  = hp_cm[b];
    float v = raw * isf + (ema[i] * cm + raw * (1.f - cm)) * (1.f - isf);
    if (n < nin) v = 0.f;                          // A_mask
    A[i] = v;
  }
}

// ---- Sinkhorn-Knopp, one 512-thread block per (b,h) slice (L2-resident) ----
// lazy pending-column divisor => 2 passes/iter + 1 final pass.
// pass2 (finalize) fuses the threshold pre-op, final mask, variance and vd.
__global__ void k_sinkhorn(float* __restrict__ W, int apply_threshold, int finalize,
                           const float* __restrict__ hp_av, float* __restrict__ vd, int nin) {
  const int s = blockIdx.x;                        // slice 0..B*H-1
  float* Wp = W + (size_t)s * N_ * N_;
  const int tid = threadIdx.x;                     // 0..511
  __shared__ float Rrec[N_];                       // reciprocal row sums
  __shared__ float Crec[N_];                       // reciprocal (pending) col sums
  const float UB = 1.0f / (N_ + 1.0f);

  // INIT: optional sparsity threshold, then relu + EPS (thread-per-column)
  for (int r = 0; r < N_; ++r) {
    float v = Wp[(size_t)r * N_ + tid];
    if (apply_threshold) v = (v > UB) ? v : 0.f;
    v = fmaxf(v, 0.f) + EPS_;
    Wp[(size_t)r * N_ + tid] = v;
  }
  Crec[tid] = 1.0f;
  __syncthreads();

  const int wv = tid >> 5, lane = tid & 31;
  for (int it = 0; it < SINK_ITERS; ++it) {
    // row sums of (W * pending colrec): wave-per-32-rows, shuffle reduce
    for (int r = wv; r < N_; r += 16) {
      float acc = 0.f;
      for (int c = lane; c < N_; c += 32)
        acc += Wp[(size_t)r * N_ + c] * Crec[c];
      acc = wave_reduce_sum(acc);
      if (lane == 0) Rrec[r] = 1.0f / acc;
    }
    __syncthreads();
    // apply pending col + new row normalization; gather new col sums
    {
      const float invc = Crec[tid];
      float acc = 0.f;
      for (int r = 0; r < N_; ++r) {
        const float v = Wp[(size_t)r * N_ + tid] * invc * Rrec[r];
        Wp[(size_t)r * N_ + tid] = v;
        acc += v;
      }
      Crec[tid] = 1.0f / acc;                      // new pending divisor
    }
    __syncthreads();
  }
  // final pending column divide (+ mask + variance on finalize)
  {
    const float invc = Crec[tid];
    float sum = 0.f, sq = 0.f;
    for (int r = 0; r < N_; ++r) {
      float v = Wp[(size_t)r * N_ + tid] * invc;
      if (finalize && r < nin) v = 0.f;            // A_mask
      Wp[(size_t)r * N_ + tid] = v;
      sum += v; sq += v * v;
    }
    if (finalize) {
      __syncthreads();
      Rrec[tid] = sum; Crec[tid] = sq;
      __syncthreads();
      for (int st = N_ / 2; st > 0; st >>= 1) {
        if (tid < st) { Rrec[tid] += Rrec[tid + st]; Crec[tid] += Crec[tid + st]; }
        __syncthreads();
      }
      if (tid == 0) {
        const float inv = 1.0f / ((float)N_ * N_);
        const float mean = Rrec[0] * inv;
        const float var  = Crec[0] * inv - mean * mean;
        vd[s] = fmaxf(hp_av[s / H_] - var, 0.f);
      }
    }
  }
}

// ---- target = softsign(A @ V) via chained accumulating WMMA (K chunks of 4) ----
__global__ void k_target(const float* __restrict__ A, const float* __restrict__ outp,
                         const float* __restrict__ noise, const float* __restrict__ vd,
                         float* __restrict__ out) {
  const int lane = threadIdx.x & 31;
  const int wid  = blockIdx.x * (blockDim.x >> 5) + (threadIdx.x >> 5);
  const int bh = wid >> 5;
  const int nt = wid & 31;
  const int b = bh >> 4, h = bh & 15;
  const int j  = lane & 15;                        // B-matrix column / A-row-within-tile
  const int k0 = (lane >> 4) << 1;
  const float vdv = vd[bh];
  const float* Ap = A + (size_t)bh * N_ * N_;
  v8f c = {};
  for (int kt = 0; kt < N_ / HD_; ++kt) {
    const int m0 = kt * HD_;
    const v2f a = *(const v2f*)(Ap + (size_t)(nt * 16 + j) * N_ + m0 + k0);
    v2f bm;
    if (j < HD_) {                                 // V occupies first 4 of 16 B columns
      const int m = m0 + k0;
      bm.x = outp[((size_t)(b * N_ + m)     * D_) + h * HD_ + j]
           + vdv * noise[((size_t)bh * N_ + m)     * HD_ + j];
      bm.y = outp[((size_t)(b * N_ + m + 1) * D_) + h * HD_ + j]
           + vdv * noise[((size_t)bh * N_ + m + 1) * HD_ + j];
    } else { bm.x = 0.f; bm.y = 0.f; }
    c = __builtin_amdgcn_wmma_f32_16x16x4_f32(false, a, false, bm, (short)0, c, false, false);
  }
  if (j < HD_) {
    #pragma unroll
    for (int r = 0; r < 8; ++r) {
      const int row = nt * 16 + ((lane >> 4) << 3) + r;
      float v = c[r];
      v = v / (1.0f + fabsf(v));                   // softsign
      out[((size_t)(b * N_ + row) * D_) + h * HD_ + j] = v;
    }
  }
}

// ---- overwrite first `nin` nodes with env input ----
__global__ void k_env(const float* __restrict__ env, float* __restrict__ out, int nin) {
  const int total = B_ * nin * D_;
  for (int i = blockIdx.x * blockDim.x + threadIdx.x; i < total;
       i += gridDim.x * blockDim.x) {
    const int b = i / (nin * D_);
    const int rem = i % (nin * D_);
    out[((size_t)(b * N_ + rem / D_) * D_) + (rem % D_)] = env[i];
  }
}

extern "C" void kernel_launch(void* const* d_in, const int* in_sizes, int n_in,
                              void* d_out, int out_size, void* d_ws, size_t ws_size,
                              hipStream_t stream) {
  const float* state        = (const float*)d_in[0];
  const float* output       = (const float*)d_in[1];
  // d_in[2] = w1 : dead in the reference (`prediction` unused) -> never read
  const float* w2           = (const float*)d_in[3];
  const float* w3           = (const float*)d_in[4];
  const float* A_ema        = (const float*)d_in[5];
  const float* hp_untanh    = (const float*)d_in[6];
  const float* hp_c_moment  = (const float*)d_in[7];
  const float* hp_a_variance= (const float*)d_in[8];
  const float* env_input    = (const float*)d_in[9];
  const float* noise        = (const float*)d_in[10];
  const float* step_counter = (const float*)d_in[11];
  const int nin = in_sizes[9] / (B_ * D_);         // num_input_nodes, host-derivable
  float* out = (float*)d_out;

  char* ws = (char*)d_ws;
  float* Q  = (float*)(ws);                                        // 512 KB
  float* K  = (float*)(ws + (size_t)524288);                       // 512 KB
  float* A  = (float*)(ws + (size_t)1048576);                      // 67.1 MB, reused in place
  float* ss = (float*)(ws + (size_t)1048576 + (size_t)B_*H_*N_*N_*4);
  float* vd = ss + B_ * H_;

  k_init    <<<1, 64, 0, stream>>>(ss);
  k_proj    <<<B_ * N_, D_, 0, stream>>>(state, w2, w3, hp_untanh, Q, K);
  k_rawA    <<<256, 256, 0, stream>>>(Q, K, A, ss);
  k_ema     <<<2048, 256, 0, stream>>>(A, A_ema, ss, hp_c_moment, step_counter, nin);
  k_sinkhorn<<<B_ * H_, N_, 0, stream>>>(A, /*threshold=*/0, /*finalize=*/0, hp_a_variance, vd, nin);
  k_sinkhorn<<<B_ * H_, N_, 0, stream>>>(A, /*threshold=*/1, /*finalize=*/1, hp_a_variance, vd, nin);
  k_target  <<<256, 256, 0, stream>>>(A, output, noise, vd, out);
  k_env     <<<64, 256, 0, stream>>>(env_input, out, nin);
}